// BaseDPS_10943576670591
// MI455X (gfx1250) — compile-verified
//
#include <hip/hip_runtime.h>
#include <hip/hip_bf16.h>
#include <math.h>

// ---------------- problem constants (match reference) ----------------
#define L_OBS      512
#define BATCH      8
#define NSEG       8192
#define EPS_PAR    1.0e-4f
#define RAYS       (BATCH * L_OBS)          // 4096
#define BLOCK_T    128                      // 4 wave32s
#define SEG_CHUNK  1024                     // 16 KB of float4 per chunk in LDS
#define NCHUNK     (NSEG / SEG_CHUNK)       // 8
#define BEAM_STEP  ((float)(6.283185307179586 / 512.0))

typedef __attribute__((ext_vector_type(2))) float v2f;
typedef __attribute__((ext_vector_type(8))) float v8f;

// ------------------------------------------------------------------
// Async-stage one chunk of segments (SEG_CHUNK float4 = 16 KB) global->LDS
// via the CDNA5 ASYNCcnt path. GVS addressing: saddr = 64-bit SGPR base,
// vaddr = 32-bit byte offset; imm offset applies to BOTH LDS & global sides.
// ------------------------------------------------------------------
__device__ __forceinline__ void stage_chunk_async(const float* __restrict__ seg,
                                                  int segBase, float4* s_seg) {
    const int tid = threadIdx.x;
    unsigned lds0 = (unsigned)(size_t)(&s_seg[tid]);          // LDS byte offset
    unsigned voff = (unsigned)(tid * 16);                     // global byte offset
    unsigned long long gbase =
        (unsigned long long)(const void*)(seg + (size_t)segBase * 4);
    asm volatile(
        "global_load_async_to_lds_b128 %0, %1, %2 offset:0\n\t"
        "global_load_async_to_lds_b128 %0, %1, %2 offset:2048\n\t"
        "global_load_async_to_lds_b128 %0, %1, %2 offset:4096\n\t"
        "global_load_async_to_lds_b128 %0, %1, %2 offset:6144\n\t"
        "global_load_async_to_lds_b128 %0, %1, %2 offset:8192\n\t"
        "global_load_async_to_lds_b128 %0, %1, %2 offset:10240\n\t"
        "global_load_async_to_lds_b128 %0, %1, %2 offset:12288\n\t"
        "global_load_async_to_lds_b128 %0, %1, %2 offset:14336\n\t"
        "s_wait_asynccnt 0x0"
        :: "v"(lds0), "v"(voff), "s"(gbase) : "memory");
    // Each lane pre-transforms exactly the entries it staged:
    // (x3,y3,x4,y4) -> (x3,y3,sx,sy).  Own-wave async already waited.
#pragma unroll
    for (int k = 0; k < SEG_CHUNK / BLOCK_T; ++k) {
        int idx = tid + k * BLOCK_T;
        float4 q = s_seg[idx];
        s_seg[idx] = make_float4(q.x, q.y, q.z - q.x, q.w - q.y);
    }
}

// One ray-vs-segment test, division-free (sign-folded cross-mul compare).
__device__ __forceinline__ void seg_test(const float4 q, float x1, float y1,
                                         float rx, float ry,
                                         float& aBest, float& rBest) {
    const float ldx = x1 - q.x;
    const float ldy = y1 - q.y;
    const float sx = q.z, sy = q.w;
    const float num_a = sx * ldy - sy * ldx;
    const float num_b = rx * ldy - ry * ldx;
    const float rxs   = sy * rx - sx * ry;
    const bool  neg   = rxs < 0.0f;
    const float r  = fabsf(rxs);
    const float a  = neg ? -num_a : num_a;
    const float bb = neg ? -num_b : num_b;
    const bool valid = (r >= EPS_PAR) & (bb >= 0.0f) & (bb <= r) & (a >= 0.0f);
    const bool better = valid && (a * rBest < aBest * r);
    aBest = better ? a : aBest;
    rBest = better ? r : rBest;
}

__device__ __forceinline__ void ray_setup(const float* __restrict__ pose, int ray,
                                          float& x1, float& y1, float& th,
                                          float& rx, float& ry) {
    const int b = ray >> 9;            // ray / 512
    const int beam = ray & 511;
    x1 = pose[b * 3 + 0];
    y1 = pose[b * 3 + 1];
    th = pose[b * 3 + 2];
    const float beamA = (float)beam * BEAM_STEP;
    const float ang = beamA + th;
    rx = cosf(ang);
    ry = sinf(ang);
}

__device__ __forceinline__ void emit_outputs(float* __restrict__ out, int ray,
                                             const float* __restrict__ seg,
                                             float x1, float y1, float th,
                                             float rx, float ry,
                                             float aBest, float rBest) {
    float u;
    if (aBest < INFINITY) {
        u = aBest / rBest;             // == num_a / rxs of winning segment
    } else {
        // reference fallback: argmin of all-inf picks index 0 -> raw u_a[0]
        const float x3 = seg[0], y3 = seg[1];
        const float sx = seg[2] - x3, sy = seg[3] - y3;
        const float rxs = sy * rx - sx * ry;
        const float num_a = sx * (y1 - y3) - sy * (x1 - x3);
        u = (fabsf(rxs) < EPS_PAR) ? 0.0f : (num_a / rxs);
    }
    const float ix = x1 + rx * u;
    const float iy = y1 + ry * u;
    const float dx = ix - x1;
    const float dy = iy - y1;
    const float c = cosf(th), s = sinf(th);
    out[ray * 2 + 0] = ix;                     // obs_global
    out[ray * 2 + 1] = iy;
    out[2 * RAYS + ray * 2 + 0] = dx * c + dy * s;   // obs_local = R^T * d
    out[2 * RAYS + ray * 2 + 1] = dy * c - dx * s;
}

// ---------------- kernel 1: WMMA partial argmin over one segment chunk ------
// One wave owns a 16-ray M-tile (16 | 512 => tile never crosses a batch).
// Per 16-segment N-tile, two v_wmma_f32_16x16x4_f32 (K=2 used) produce the
// full 16x16 rxs and num_b tiles; validity/argmin runs on the C layout
// (lane = column segment, VGPR index = row ray), then a 4-step shfl_xor
// min-reduction folds the 16 columns per half-wave.
__global__ __launch_bounds__(BLOCK_T) void ray_partial_wmma(
        const float* __restrict__ seg, const float* __restrict__ pose,
        float2* __restrict__ part) {
    __shared__ float4 s_seg[SEG_CHUNK];
    const int chunk = blockIdx.y;
    stage_chunk_async(seg, chunk * SEG_CHUNK, s_seg);
    __syncthreads();

    const int lane    = threadIdx.x & 31;
    const int wave    = threadIdx.x >> 5;
    const int rayBase = blockIdx.x * 64 + wave * 16;   // whole block same batch

    float x1, y1, th, rxv, ryv;
    ray_setup(pose, rayBase + (lane & 15), x1, y1, th, rxv, ryv);

#if defined(__AMDGCN__) && __has_builtin(__builtin_amdgcn_wmma_f32_16x16x4_f32)
    const bool hi = lane >= 16;
    // A (16x4): row m = [rx_m, -ry_m, 0, 0]; lanes16-31 carry K=2,3 = 0.
    v2f A;
    A[0] = hi ? 0.0f : rxv;
    A[1] = hi ? 0.0f : -ryv;

    float aBest[8], rBest[8];
#pragma unroll
    for (int r = 0; r < 8; ++r) { aBest[r] = INFINITY; rBest[r] = 1.0f; }

    for (int tb = 0; tb < SEG_CHUNK; tb += 16) {
        const float4 q = s_seg[tb + (lane & 15)];      // this lane's column seg
        const float ldx = x1 - q.x;
        const float ldy = y1 - q.y;
        const float num_a = q.z * ldy - q.w * ldx;     // column-constant
        // B (4x16): col n = [sy_n, sx_n, 0, 0]  /  [ldy_n, ldx_n, 0, 0]
        v2f Br, Bn;
        Br[0] = hi ? 0.0f : q.w;   // sy
        Br[1] = hi ? 0.0f : q.z;   // sx
        Bn[0] = hi ? 0.0f : ldy;
        Bn[1] = hi ? 0.0f : ldx;
        v8f Crxs = {}, Cnumb = {};
        Crxs = __builtin_amdgcn_wmma_f32_16x16x4_f32(
            false, A, false, Br, (short)0, Crxs, false, false);
        Cnumb = __builtin_amdgcn_wmma_f32_16x16x4_f32(
            false, A, false, Bn, (short)0, Cnumb, false, false);
#pragma unroll
        for (int r = 0; r < 8; ++r) {
            const float rxs  = Crxs[r];
            const float numb = Cnumb[r];
            const bool  neg  = rxs < 0.0f;
            const float rr = fabsf(rxs);
            const float aa = neg ? -num_a : num_a;
            const float bb = neg ? -numb : numb;
            const bool valid =
                (rr >= EPS_PAR) & (bb >= 0.0f) & (bb <= rr) & (aa >= 0.0f);
            const bool better = valid && (aa * rBest[r] < aBest[r] * rr);
            aBest[r] = better ? aa : aBest[r];
            rBest[r] = better ? rr : rBest[r];
        }
    }
    // Fold 16 columns per half-wave (masks < 16 stay inside each half).
#pragma unroll
    for (int m = 8; m >= 1; m >>= 1) {
#pragma unroll
        for (int r = 0; r < 8; ++r) {
            const float oa = __shfl_xor(aBest[r], m, 32);
            const float orr = __shfl_xor(rBest[r], m, 32);
            const bool take = (oa * rBest[r]) < (aBest[r] * orr);
            aBest[r] = take ? oa : aBest[r];
            rBest[r] = take ? orr : rBest[r];
        }
    }
    if ((lane & 15) == 0) {                 // lane0 -> rows 0-7, lane16 -> 8-15
        const int base = rayBase + (hi ? 8 : 0);
#pragma unroll
        for (int r = 0; r < 8; ++r)
            part[(size_t)(base + r) * NCHUNK + chunk] =
                make_float2(aBest[r], rBest[r]);
    }
#else
    // Builtin unavailable: scalar degrade, lanes 0-15 own one ray each.
    if (lane < 16) {
        float aBest = INFINITY, rBest = 1.0f;
        for (int j = 0; j < SEG_CHUNK; ++j)
            seg_test(s_seg[j], x1, y1, rxv, ryv, aBest, rBest);
        part[(size_t)(rayBase + lane) * NCHUNK + chunk] =
            make_float2(aBest, rBest);
    }
#endif
}

// ---------------- kernel 2: fold chunk partials, emit outputs ---------------
__global__ __launch_bounds__(BLOCK_T) void ray_final(
        const float* __restrict__ seg, const float* __restrict__ pose,
        const float2* __restrict__ part, float* __restrict__ out) {
    const int ray = blockIdx.x * BLOCK_T + threadIdx.x;
    float x1, y1, th, rx, ry;
    ray_setup(pose, ray, x1, y1, th, rx, ry);

    float aBest = INFINITY, rBest = 1.0f;
#pragma unroll
    for (int c = 0; c < NCHUNK; ++c) {          // ascending order keeps the
        const float2 p = part[(size_t)ray * NCHUNK + c];   // first-occurrence min
        const bool better = (p.x * rBest) < (aBest * p.y);
        aBest = better ? p.x : aBest;
        rBest = better ? p.y : rBest;
    }
    emit_outputs(out, ray, seg, x1, y1, th, rx, ry, aBest, rBest);
}

// ---------------- fallback: monolithic (no workspace needed) ----------------
__global__ __launch_bounds__(BLOCK_T) void ray_mono(
        const float* __restrict__ seg, const float* __restrict__ pose,
        float* __restrict__ out) {
    __shared__ float4 s_seg[SEG_CHUNK];
    const int ray = blockIdx.x * BLOCK_T + threadIdx.x;
    float x1, y1, th, rx, ry;
    ray_setup(pose, ray, x1, y1, th, rx, ry);

    float aBest = INFINITY, rBest = 1.0f;
    for (int chunk = 0; chunk < NCHUNK; ++chunk) {
        __syncthreads();                         // readers of prev chunk done
        stage_chunk_async(seg, chunk * SEG_CHUNK, s_seg);
        __syncthreads();                         // all waves staged+transformed
        for (int j = 0; j < SEG_CHUNK; ++j)
            seg_test(s_seg[j], x1, y1, rx, ry, aBest, rBest);
    }
    emit_outputs(out, ray, seg, x1, y1, th, rx, ry, aBest, rBest);
}

// ---------------------------------------------------------------------------
extern "C" void kernel_launch(void* const* d_in, const int* in_sizes, int n_in,
                              void* d_out, int out_size, void* d_ws, size_t ws_size,
                              hipStream_t stream) {
    const float* seg  = (const float*)d_in[0];   // line_seg [8192,4] f32
    const float* pose = (const float*)d_in[1];   // pose     [8,3]    f32
    float* out = (float*)d_out;                  // 16384 f32 (global ++ local)

    const size_t need = (size_t)RAYS * NCHUNK * sizeof(float2);  // 256 KB
    if (d_ws != nullptr && ws_size >= need) {
        float2* part = (float2*)d_ws;
        ray_partial_wmma<<<dim3(RAYS / 64, NCHUNK), BLOCK_T, 0, stream>>>(seg, pose, part);
        ray_final<<<RAYS / BLOCK_T, BLOCK_T, 0, stream>>>(seg, pose, part, out);
    } else {
        ray_mono<<<RAYS / BLOCK_T, BLOCK_T, 0, stream>>>(seg, pose, out);
    }
}